// Attention_2637109920361
// MI455X (gfx1250) — compile-verified
//
#include <hip/hip_runtime.h>
#include <hip/hip_bf16.h>
#include <math.h>

// Problem constants (from reference): L=1024, B=64, DEC=512, ALIGN=512, DENC=1024
#define L_DIM 1024
#define B_DIM 64
#define DEC_DIM 512
#define ALIGN_DIM 512
#define DENC_DIM 1024

typedef __attribute__((ext_vector_type(2))) float v2f;
typedef __attribute__((ext_vector_type(8))) float v8f;

// ---------------------------------------------------------------------------
// Kernel 0: zero attend output region, copy a1_w into its output slot
// ---------------------------------------------------------------------------
__global__ void misc_init_kernel(const float* __restrict__ a1_w,
                                 float* __restrict__ out_a1w,
                                 float* __restrict__ out_attend) {
    int idx = blockIdx.x * 256 + threadIdx.x;
    if (idx < B_DIM * DENC_DIM) out_attend[idx] = 0.0f;
    if (idx < ALIGN_DIM)        out_a1w[idx]   = a1_w[idx];
}

// ---------------------------------------------------------------------------
// Kernel 1: sa_s[b,a] = sum_k s_tm1[b,k] * sa_w[a,k] + sa_b[a]
// One wave per 16x16 output tile; K swept 4 at a time with
// V_WMMA_F32_16X16X4_F32 (exact fp32 matrix path on CDNA5).
//
// ISA 32-bit A 16x4 layout: lanes 0-15 -> M=lane (K pair 0,1 in VGPR0/1),
// lanes 16-31 -> M=lane-16 (K pair 2,3). B 4x16 mirrors with N across lanes.
// Both become a contiguous float2 load per lane.
// ---------------------------------------------------------------------------
__global__ void sa_gemm_wmma_kernel(const float* __restrict__ s_tm1,
                                    const float* __restrict__ sa_w,
                                    const float* __restrict__ sa_b,
                                    float* __restrict__ sa_s) {
    const int wave = (blockIdx.x * blockDim.x + threadIdx.x) >> 5;
    const int lane = threadIdx.x & 31;
    const int bt = wave >> 5;          // 0..3   (B_DIM/16 tiles of M)
    const int at = wave & 31;          // 0..31  (ALIGN_DIM/16 tiles of N)

    const int mn    = lane & 15;       // row within tile (M for A, N for B)
    const int khalf = (lane >> 4) * 2; // 0 or 2: which K pair this half-wave holds

    const float* aptr = s_tm1 + (bt * 16 + mn) * DEC_DIM + khalf;
    const float* bptr = sa_w  + (at * 16 + mn) * DEC_DIM + khalf; // B[k,n] = sa_w[n,k]

    v8f c = {};
#pragma unroll 8
    for (int k0 = 0; k0 < DEC_DIM; k0 += 4) {
        v2f a = *(const v2f*)(aptr + k0);
        v2f b = *(const v2f*)(bptr + k0);
        // (neg_a, A, neg_b, B, c_mod, C, reuse_a, reuse_b)
        c = __builtin_amdgcn_wmma_f32_16x16x4_f32(false, a, false, b,
                                                  (short)0, c, false, false);
    }

    // D layout: VGPR r holds M=r (lanes 0-15) / M=r+8 (lanes 16-31), N=lane&15
    const int n    = lane & 15;
    const int moff = (lane >> 4) * 8;
    const float bias = sa_b[at * 16 + n];
#pragma unroll
    for (int r = 0; r < 8; ++r) {
        sa_s[(bt * 16 + moff + r) * ALIGN_DIM + at * 16 + n] = c[r] + bias;
    }
}

// ---------------------------------------------------------------------------
// Kernel 2: streaming fused tanh + a1 dot. One wave per (l,b) row:
//   tanh_sa[l,b,:] = tanh(sa_s[b,:] + uh[l,b,:])   (128MB in, 128MB out)
//   a1[l,b] = dot(tanh_sa[l,b,:], a1_w) + a1_b
// sa_s (128KB) and a1_w (2KB) stay L2/L0-resident.
// ---------------------------------------------------------------------------
__global__ void tanh_a1_kernel(const float* __restrict__ uh,
                               const float* __restrict__ sa_s,
                               const float* __restrict__ a1_w,
                               const float* __restrict__ a1_b,
                               float* __restrict__ tanh_out,
                               float* __restrict__ a1_out) {
    const int wavesPerBlock = blockDim.x >> 5;
    const int row  = blockIdx.x * wavesPerBlock + (threadIdx.x >> 5); // l*B + b
    const int lane = threadIdx.x & 31;
    const int b    = row & (B_DIM - 1);

    const float* uhrow = uh + (size_t)row * ALIGN_DIM;
    const float* srow  = sa_s + b * ALIGN_DIM;
    float*       trow  = tanh_out + (size_t)row * ALIGN_DIM;

    float acc = 0.0f;
#pragma unroll
    for (int i = lane * 4; i < ALIGN_DIM; i += 32 * 4) {
        float4 u = *(const float4*)(uhrow + i);
        float4 s = *(const float4*)(srow + i);
        float4 w = *(const float4*)(a1_w + i);
        float4 t;
        t.x = tanhf(u.x + s.x);
        t.y = tanhf(u.y + s.y);
        t.z = tanhf(u.z + s.z);
        t.w = tanhf(u.w + s.w);
        *(float4*)(trow + i) = t;
        acc += t.x * w.x + t.y * w.y + t.z * w.z + t.w * w.w;
    }
    // wave32 reduction
#pragma unroll
    for (int off = 16; off > 0; off >>= 1)
        acc += __shfl_down(acc, off, 32);
    if (lane == 0) a1_out[row] = acc + a1_b[0];
}

// ---------------------------------------------------------------------------
// Kernel 3: masked softmax over L per column b (reference semantics:
// max unmasked, e = exp(x-max)*mask, out = e / sum(e)).
// One 256-thread block per b; LDS tree reductions.
// ---------------------------------------------------------------------------
__global__ void softmax_kernel(const float* __restrict__ a1,
                               const float* __restrict__ xs_mask,
                               float* __restrict__ e_out) {
    const int b = blockIdx.x;
    __shared__ float red[256];

    float m = -INFINITY;
    for (int l = threadIdx.x; l < L_DIM; l += 256)
        m = fmaxf(m, a1[l * B_DIM + b]);
    red[threadIdx.x] = m;
    __syncthreads();
    for (int s = 128; s > 0; s >>= 1) {
        if (threadIdx.x < s)
            red[threadIdx.x] = fmaxf(red[threadIdx.x], red[threadIdx.x + s]);
        __syncthreads();
    }
    m = red[0];
    __syncthreads();

    float vals[L_DIM / 256];
    float sum = 0.0f;
#pragma unroll
    for (int i = 0; i < L_DIM / 256; ++i) {
        int l = threadIdx.x + i * 256;
        float ex = __expf(a1[l * B_DIM + b] - m) * xs_mask[l * B_DIM + b];
        vals[i] = ex;
        sum += ex;
    }
    red[threadIdx.x] = sum;
    __syncthreads();
    for (int s = 128; s > 0; s >>= 1) {
        if (threadIdx.x < s)
            red[threadIdx.x] += red[threadIdx.x + s];
        __syncthreads();
    }
    const float inv = 1.0f / red[0];
#pragma unroll
    for (int i = 0; i < L_DIM / 256; ++i) {
        int l = threadIdx.x + i * 256;
        e_out[l * B_DIM + b] = vals[i] * inv;
    }
}

// ---------------------------------------------------------------------------
// Kernel 4: attend[b,d] = sum_l e[l,b] * xs_h[l,b,d]  (256MB stream of xs_h)
// grid = (b=64, lchunk=16); block 256; thread owns 4 contiguous d (float4),
// so each l-iteration is a fully coalesced 4KB row read. Partials combined
// with global_atomic_add_f32.
// ---------------------------------------------------------------------------
__global__ void attend_kernel(const float* __restrict__ e_ij,
                              const float* __restrict__ xs_h,
                              float* __restrict__ attend) {
    const int b  = blockIdx.x;          // 0..63
    const int l0 = blockIdx.y * 64;     // 16 chunks of 64
    __shared__ float es[64];
    if (threadIdx.x < 64)
        es[threadIdx.x] = e_ij[(l0 + threadIdx.x) * B_DIM + b];
    __syncthreads();

    const int d = threadIdx.x * 4;
    const float* base = xs_h + ((size_t)l0 * B_DIM + b) * DENC_DIM + d;
    float4 acc = make_float4(0.f, 0.f, 0.f, 0.f);
#pragma unroll 4
    for (int i = 0; i < 64; ++i) {
        float4 x = *(const float4*)(base + (size_t)i * (B_DIM * DENC_DIM));
        const float w = es[i];
        acc.x += w * x.x;
        acc.y += w * x.y;
        acc.z += w * x.z;
        acc.w += w * x.w;
    }
    float* o = attend + b * DENC_DIM + d;
    atomicAdd(o + 0, acc.x);
    atomicAdd(o + 1, acc.y);
    atomicAdd(o + 2, acc.z);
    atomicAdd(o + 3, acc.w);
}

// ---------------------------------------------------------------------------
// Host launcher
// ---------------------------------------------------------------------------
extern "C" void kernel_launch(void* const* d_in, const int* in_sizes, int n_in,
                              void* d_out, int out_size, void* d_ws, size_t ws_size,
                              hipStream_t stream) {
    (void)in_sizes; (void)n_in; (void)out_size; (void)ws_size;

    const float* s_tm1   = (const float*)d_in[0]; // [B, DEC]
    const float* xs_h    = (const float*)d_in[1]; // [L, B, DENC]
    const float* uh      = (const float*)d_in[2]; // [L, B, ALIGN]
    const float* xs_mask = (const float*)d_in[3]; // [L, B]
    const float* sa_w    = (const float*)d_in[4]; // [ALIGN, DEC]
    const float* sa_b    = (const float*)d_in[5]; // [ALIGN]
    const float* a1_w    = (const float*)d_in[6]; // [1, ALIGN]
    const float* a1_b    = (const float*)d_in[7]; // [1]

    // d_out layout: tanh_sa | a1_w | a1 | e_ij | attend
    float* out        = (float*)d_out;
    float* out_tanh   = out;                                     // L*B*ALIGN
    float* out_a1w    = out_tanh + (size_t)L_DIM * B_DIM * ALIGN_DIM;
    float* out_a1     = out_a1w + ALIGN_DIM;                     // L*B
    float* out_e      = out_a1 + L_DIM * B_DIM;                  // L*B
    float* out_attend = out_e + L_DIM * B_DIM;                   // B*DENC

    float* sa_s = (float*)d_ws;                                  // B*ALIGN fp32

    // 0) init attend=0, copy a1_w out
    misc_init_kernel<<<(B_DIM * DENC_DIM + 255) / 256, 256, 0, stream>>>(
        a1_w, out_a1w, out_attend);

    // 1) sa_s = s_tm1 @ sa_w.T + sa_b  via WMMA f32 16x16x4
    //    128 tiles -> 128 waves -> 16 blocks x 8 waves
    sa_gemm_wmma_kernel<<<16, 256, 0, stream>>>(s_tm1, sa_w, sa_b, sa_s);

    // 2) tanh + a1 dot: one wave per (l,b) row, 8 rows per block
    tanh_a1_kernel<<<(L_DIM * B_DIM) / 8, 256, 0, stream>>>(
        uh, sa_s, a1_w, a1_b, out_tanh, out_a1);

    // 3) masked softmax over L, per b
    softmax_kernel<<<B_DIM, 256, 0, stream>>>(out_a1, xs_mask, out_e);

    // 4) attend accumulation over L chunks
    attend_kernel<<<dim3(B_DIM, L_DIM / 64), 256, 0, stream>>>(
        out_e, xs_h, out_attend);
}